// MPJPELoss_75093208203681
// MI455X (gfx1250) — compile-verified
//
#include <hip/hip_runtime.h>
#include <stdint.h>

// MPJPE loss, MI455X (gfx1250, wave32).
// Memory-bound: 402 MB streamed once -> ~17us floor at 23.3 TB/s.
// Strategy: double-buffered GLOBAL_LOAD_ASYNC_TO_LDS_B128 streaming
// (ASYNCcnt-synchronized, per-wave self-contained so no barriers in the
// hot loop), wave32 ds_swizzle XOR-butterfly reduction, deterministic
// two-pass final reduction.

#define GRID1   2048
#define BLOCK   256
#define NWAVES  (BLOCK / 32)
#define ITERS   8
// BS=262144 samples * 16 groups-of-4-joints = 4,194,304 groups total.
// GRID1*BLOCK = 524,288 threads -> exactly 8 groups per thread.
#define STRIDE  ((size_t)GRID1 * BLOCK)
#define INV_COUNT (1.0f / 16515072.0f)   // 262144 * 63 counted joints

// ---- CDNA5 async global->LDS copy of 16 bytes (per lane) ----------------
__device__ __forceinline__ void async_copy16(uint32_t lds_addr, uint64_t gaddr) {
  asm volatile("global_load_async_to_lds_b128 %0, %1, off"
               :: "v"(lds_addr), "v"(gaddr)
               : "memory");
}

// ---- per-joint ratio ||p-q|| / ||q|| ------------------------------------
__device__ __forceinline__ float joint_ratio(float px, float py, float pz,
                                             float qx, float qy, float qz) {
  float dx = px - qx, dy = py - qy, dz = pz - qz;
  float dd = dx * dx + dy * dy + dz * dz;
  float qq = qx * qx + qy * qy + qz * qz;
  return __builtin_amdgcn_sqrtf(dd) * __builtin_amdgcn_rsqf(qq);
}

// ---- wave32 XOR-butterfly sum via ds_swizzle (group-of-32 mode) ---------
__device__ __forceinline__ float wave_reduce_add(float v) {
  v += __int_as_float(__builtin_amdgcn_ds_swizzle(__float_as_int(v), 0x401F)); // xor 16
  v += __int_as_float(__builtin_amdgcn_ds_swizzle(__float_as_int(v), 0x201F)); // xor 8
  v += __int_as_float(__builtin_amdgcn_ds_swizzle(__float_as_int(v), 0x101F)); // xor 4
  v += __int_as_float(__builtin_amdgcn_ds_swizzle(__float_as_int(v), 0x081F)); // xor 2
  v += __int_as_float(__builtin_amdgcn_ds_swizzle(__float_as_int(v), 0x041F)); // xor 1
  return v;
}

// =========================================================================
// Pass 1: stream all joints, one block-partial sum per block into d_ws.
// Each lane owns a group of 4 joints (48 B) per tensor per iteration,
// staged through LDS by async b128 loads, double-buffered.
// =========================================================================
__global__ void __launch_bounds__(BLOCK) mpjpe_pass1(
    const float4* __restrict__ pred, const float4* __restrict__ gt,
    float* __restrict__ partial) {
  __shared__ float4 tile[2][2][BLOCK][3];   // [buf][pred/gt][lane][3xb128] = 48 KB
  __shared__ float  wsum[NWAVES];

  const int    tid = threadIdx.x;
  const size_t t0  = (size_t)blockIdx.x * BLOCK + tid;
  // group index g = t0 + it*STRIDE; STRIDE % 16 == 0, so g%16 == tid%16.
  // Group with g%16==0 starts at joint 0 of a sample -> exclude joint A.
  const bool haveA = (tid & 15) != 0;

  // LDS byte addresses of this lane's slots (flat-pointer low 32 bits are
  // the LDS byte address per the gfx1250 aperture mapping).
  uint32_t lp[2], lg[2];
  lp[0] = (uint32_t)(uintptr_t)&tile[0][0][tid][0];
  lg[0] = (uint32_t)(uintptr_t)&tile[0][1][tid][0];
  lp[1] = (uint32_t)(uintptr_t)&tile[1][0][tid][0];
  lg[1] = (uint32_t)(uintptr_t)&tile[1][1][tid][0];

  auto issue = [&](int buf, size_t g) {
    uint64_t pa = (uint64_t)(uintptr_t)(pred + g * 3);
    uint64_t ga = (uint64_t)(uintptr_t)(gt + g * 3);
    async_copy16(lp[buf],      pa);
    async_copy16(lp[buf] + 16, pa + 16);
    async_copy16(lp[buf] + 32, pa + 32);
    async_copy16(lg[buf],      ga);
    async_copy16(lg[buf] + 16, ga + 16);
    async_copy16(lg[buf] + 32, ga + 32);
  };

  issue(0, t0);                       // prime buffer 0
  float sum = 0.0f;

  for (int it = 0; it < ITERS; ++it) {
    const int b = it & 1;
    if (it + 1 < ITERS) {
      issue((it + 1) & 1, t0 + (size_t)(it + 1) * STRIDE);
      // 12 async ops outstanding; <=6 left means current buffer landed.
      asm volatile("s_wait_asynccnt 0x6" ::: "memory");
    } else {
      asm volatile("s_wait_asynccnt 0x0" ::: "memory");
    }

    const float4 p0 = tile[b][0][tid][0];
    const float4 p1 = tile[b][0][tid][1];
    const float4 p2 = tile[b][0][tid][2];
    const float4 q0 = tile[b][1][tid][0];
    const float4 q1 = tile[b][1][tid][1];
    const float4 q2 = tile[b][1][tid][2];

    float rA = joint_ratio(p0.x, p0.y, p0.z, q0.x, q0.y, q0.z);
    float rB = joint_ratio(p0.w, p1.x, p1.y, q0.w, q1.x, q1.y);
    float rC = joint_ratio(p1.z, p1.w, p2.x, q1.z, q1.w, q2.x);
    float rD = joint_ratio(p2.y, p2.z, p2.w, q2.y, q2.z, q2.w);

    // select (not multiply): keeps a potential 0/0 NaN at joint 0 out.
    sum += (haveA ? rA : 0.0f) + rB + rC + rD;
  }

  float w = wave_reduce_add(sum);
  if ((tid & 31) == 0) wsum[tid >> 5] = w;
  __syncthreads();
  if (tid < 32) {
    float s = (tid < NWAVES) ? wsum[tid] : 0.0f;
    s += __int_as_float(__builtin_amdgcn_ds_swizzle(__float_as_int(s), 0x101F));
    s += __int_as_float(__builtin_amdgcn_ds_swizzle(__float_as_int(s), 0x081F));
    s += __int_as_float(__builtin_amdgcn_ds_swizzle(__float_as_int(s), 0x041F));
    if (tid == 0) partial[blockIdx.x] = s;
  }
}

// =========================================================================
// Pass 2: single block deterministically reduces GRID1 partials -> mean.
// =========================================================================
__global__ void __launch_bounds__(BLOCK) mpjpe_pass2(
    const float* __restrict__ partial, float* __restrict__ out) {
  __shared__ float wsum[NWAVES];
  const int tid = threadIdx.x;

  float s = 0.0f;
  for (int i = tid; i < GRID1; i += BLOCK) s += partial[i];

  float w = wave_reduce_add(s);
  if ((tid & 31) == 0) wsum[tid >> 5] = w;
  __syncthreads();
  if (tid < 32) {
    float v = (tid < NWAVES) ? wsum[tid] : 0.0f;
    v += __int_as_float(__builtin_amdgcn_ds_swizzle(__float_as_int(v), 0x101F));
    v += __int_as_float(__builtin_amdgcn_ds_swizzle(__float_as_int(v), 0x081F));
    v += __int_as_float(__builtin_amdgcn_ds_swizzle(__float_as_int(v), 0x041F));
    if (tid == 0) out[0] = v * INV_COUNT;
  }
}

extern "C" void kernel_launch(void* const* d_in, const int* in_sizes, int n_in,
                              void* d_out, int out_size, void* d_ws, size_t ws_size,
                              hipStream_t stream) {
  (void)in_sizes; (void)n_in; (void)out_size; (void)ws_size;
  const float4* pred = (const float4*)d_in[0];
  const float4* gt   = (const float4*)d_in[1];
  float* partial     = (float*)d_ws;          // GRID1 floats = 8 KB scratch
  float* out         = (float*)d_out;

  mpjpe_pass1<<<GRID1, BLOCK, 0, stream>>>(pred, gt, partial);
  mpjpe_pass2<<<1, BLOCK, 0, stream>>>(partial, out);
}